// RoiPooling_25950192402926
// MI455X (gfx1250) — compile-verified
//
#include <hip/hip_runtime.h>
#include <hip/hip_bf16.h>

// Problem constants (from reference)
#define C_   256
#define H_   128
#define W_   128
#define HW_  (H_ * W_)
#define L_   21
#define R_   32
#define K_   160   // R * (1 + 4)

// d_out layout (floats, concatenated in reference return order)
#define OFF_POOL 0                 // K*L = 3360
#define OFF_LAB  (K_ * L_)         // 3360 .. +160
#define OFF_FG   (OFF_LAB + K_)    // 3520 .. +160
#define OFF_CONF (OFF_FG + K_)     // 3680 .. +160
#define OFF_LOC  (OFF_CONF + K_)   // 3840 .. +640  (total 4480)

typedef __attribute__((ext_vector_type(2))) float v2f;
typedef __attribute__((ext_vector_type(8))) float v8f;

// ---------------------------------------------------------------------------
// Kernel 1: patch geometry. Order: p=1 block (k in [0,32): r=k, i=0), then
// p=4 block (k in [32,160): i = (k-32)/32, r = (k-32)%32), i.e. i-outer/r-inner.
// ---------------------------------------------------------------------------
__global__ void geom_kernel(const int* __restrict__ roi,
                            const int* __restrict__ label_list,
                            int* __restrict__ geo, int* __restrict__ labs,
                            float* __restrict__ out)
{
    int k = blockIdx.x * blockDim.x + threadIdx.x;
    if (k >= K_) return;

    int r, ix, iy; float ppe;
    if (k < R_) { r = k; ix = 0; iy = 0; ppe = 1.0f; }
    else {
        int kk = k - R_;
        int i  = kk / R_;
        r  = kk % R_;
        ix = i & 1;
        iy = i >> 1;
        ppe = 2.0f;
    }
    float xmin = (float)roi[r * 4 + 0];
    float ymin = (float)roi[r * 4 + 1];
    float xmax = (float)roi[r * 4 + 2];
    float ymax = (float)roi[r * 4 + 3];
    float w = (xmax - xmin) / ppe;
    float h = (ymax - ymin) / ppe;

    int x0 = (int)floorf(xmin + (float)ix * w);
    int y0 = (int)floorf(ymin + (float)iy * h);
    int x1 = (int)ceilf (xmin + (float)(ix + 1) * w);
    int y1 = (int)ceilf (ymin + (float)(iy + 1) * h);

    geo[k * 4 + 0] = x0;
    geo[k * 4 + 1] = y0;
    geo[k * 4 + 2] = x1;
    geo[k * 4 + 3] = y1;
    int lab = label_list[r];
    labs[k] = lab;

    out[OFF_LAB + k] = (float)lab;
    out[OFF_LOC + k * 4 + 0] = (float)x0;
    out[OFF_LOC + k * 4 + 1] = (float)y0;
    out[OFF_LOC + k * 4 + 2] = (float)x1;
    out[OFF_LOC + k * 4 + 3] = (float)y1;
}

// ---------------------------------------------------------------------------
// Kernel 2: per-pixel argmax over L CAM planes (first-max, like jnp.argmax),
// plus the max value (== norm_cam_bg[pred] at that pixel).
// ---------------------------------------------------------------------------
__global__ void cam_argmax_kernel(const float* __restrict__ ncb,
                                  int* __restrict__ pred,
                                  float* __restrict__ maxv)
{
    int p = blockIdx.x * blockDim.x + threadIdx.x;
    if (p >= HW_) return;
    float best = ncb[p];
    int   bl   = 0;
    #pragma unroll
    for (int l = 1; l < L_; ++l) {
        float v = ncb[l * HW_ + p];
        if (v > best) { best = v; bl = l; }
    }
    pred[p] = bl;
    maxv[p] = best;
}

// ---------------------------------------------------------------------------
// Kernel 3: per-patch class histogram + masked CAM sums via LDS atomics.
// Produces fg_score and confidence_score.
// ---------------------------------------------------------------------------
__global__ void rect_stats_kernel(const int* __restrict__ geo,
                                  const int* __restrict__ labs,
                                  const int* __restrict__ pred,
                                  const float* __restrict__ maxv,
                                  float* __restrict__ out)
{
    int k = blockIdx.x;
    int x0 = geo[k * 4 + 0], y0 = geo[k * 4 + 1];
    int x1 = geo[k * 4 + 2], y1 = geo[k * 4 + 3];
    int wpx = x1 - x0, hpx = y1 - y0;
    int npix = wpx * hpx;

    __shared__ int   cnt_sh[L_];
    __shared__ float s_sh[L_];
    int t = threadIdx.x;
    if (t < L_) { cnt_sh[t] = 0; s_sh[t] = 0.0f; }
    __syncthreads();

    for (int i = t; i < npix; i += blockDim.x) {
        int y = y0 + i / wpx;
        int x = x0 + i % wpx;
        int p = y * W_ + x;
        int l = pred[p];
        atomicAdd(&cnt_sh[l], 1);
        atomicAdd(&s_sh[l], maxv[p]);
    }
    __syncthreads();

    if (t == 0) {
        float area = (float)npix;
        out[OFF_FG + k] = (float)cnt_sh[labs[k]] / area;
        float s = 0.0f, np = 0.0f;
        for (int l = 0; l < L_; ++l) {
            if (cnt_sh[l] > 0) {
                s  += s_sh[l] / (float)cnt_sh[l];
                np += 1.0f;
            }
        }
        out[OFF_CONF + k] = s / np;
    }
}

// ---------------------------------------------------------------------------
// Kernel 4: mf[k][c] = mean of feature_map[c] over patch rect.
// One block per patch, one thread per channel (C == 256 == blockDim).
// ---------------------------------------------------------------------------
__global__ void mf_kernel(const float* __restrict__ fm,
                          const int* __restrict__ geo,
                          float* __restrict__ mf)
{
    int k = blockIdx.x;
    int c = threadIdx.x;
    int x0 = geo[k * 4 + 0], y0 = geo[k * 4 + 1];
    int x1 = geo[k * 4 + 2], y1 = geo[k * 4 + 3];
    float inv_area = 1.0f / (float)((x1 - x0) * (y1 - y0));

    const float* __restrict__ f = fm + (size_t)c * HW_;
    float acc = 0.0f;
    for (int y = y0; y < y1; ++y) {
        const float* __restrict__ row = f + y * W_;
        for (int x = x0; x < x1; ++x) acc += row[x];
    }
    mf[k * C_ + c] = acc * inv_area;
}

// ---------------------------------------------------------------------------
// Kernel 5: pool = mf (160x256) @ cls_w^T (256x21) + b, via fp32 WMMA
// V_WMMA_F32_16X16X4_F32. One wave per 16x16 output tile; 10 M-tiles x
// 2 N-tiles = 20 blocks of 32 threads. Kdim = 256 -> 64 WMMA steps.
// Fragment layouts (ISA 7.12.2, 32-bit):
//   A 16x4: lane = m + 16*(kq), VGPR v -> K = 2*kq + v
//   B 4x16: lane = n + 16*(kq), VGPR v -> K = 2*kq + v
//   D 16x16: lane = n + 16*(mh), VGPR v -> M = v + 8*mh
// ---------------------------------------------------------------------------
__global__ void __launch_bounds__(32)
pool_wmma_kernel(const float* __restrict__ mf,
                 const float* __restrict__ wmat,   // cls_w: (L, C) row-major
                 const float* __restrict__ bias,   // cls_b: (L,)
                 float* __restrict__ out)
{
    const int MT = K_ / 16;                 // 10 M-tiles
    int bid  = blockIdx.x;
    int m0   = (bid % MT) * 16;
    int n0   = (bid / MT) * 16;

    int lane = threadIdx.x;                 // 0..31, wave32
    int half = lane >> 4;                   // K-quad / M-half selector
    int mr   = lane & 15;

    int m = m0 + mr;                        // A row for this lane
    int n = n0 + mr;                        // B column for this lane
    // Pad N beyond L with zeros (B columns 21..31 contribute nothing).
    float nmask = (n < L_) ? 1.0f : 0.0f;
    int   nc    = (n < L_) ? n : 0;

    v8f acc = {};
    for (int kk = 0; kk < C_; kk += 4) {
        int ka = kk + 2 * half;
        v2f a, b;
        a.x = mf[m * C_ + ka];
        a.y = mf[m * C_ + ka + 1];
        b.x = wmat[nc * C_ + ka]     * nmask;   // B[k][n] = cls_w[n][k]
        b.y = wmat[nc * C_ + ka + 1] * nmask;
        acc = __builtin_amdgcn_wmma_f32_16x16x4_f32(
            /*neg_a=*/false, a, /*neg_b=*/false, b,
            /*c_mod=*/(short)0, acc, /*reuse_a=*/false, /*reuse_b=*/false);
    }

    int col = n0 + mr;
    if (col < L_) {
        float bv = bias[col];
        #pragma unroll
        for (int v = 0; v < 8; ++v) {
            int row = m0 + v + 8 * half;
            out[OFF_POOL + row * L_ + col] = acc[v] + bv;
        }
    }
}

// ---------------------------------------------------------------------------
extern "C" void kernel_launch(void* const* d_in, const int* in_sizes, int n_in,
                              void* d_out, int out_size, void* d_ws, size_t ws_size,
                              hipStream_t stream)
{
    const float* fm         = (const float*)d_in[0];  // (C,H,W)
    const int*   roi        = (const int*)  d_in[1];  // (R,4)
    const int*   label_list = (const int*)  d_in[2];  // (R,)
    // d_in[3] = patch_nums (1,4) -- hardcoded
    const float* ncb        = (const float*)d_in[4];  // (L,H,W)
    // d_in[5] = cls_label -- unused by reference
    const float* cls_w      = (const float*)d_in[6];  // (L,C)
    const float* cls_b      = (const float*)d_in[7];  // (L,)
    float* out = (float*)d_out;

    // Workspace carve-up (~300 KB)
    int*   geo  = (int*)d_ws;              // K*4 ints
    int*   labs = geo + K_ * 4;            // K ints
    int*   pred = labs + K_;               // HW ints
    float* maxv = (float*)(pred + HW_);    // HW floats
    float* mf   = maxv + HW_;              // K*C floats

    geom_kernel      <<<1, 256, 0, stream>>>(roi, label_list, geo, labs, out);
    cam_argmax_kernel<<<HW_ / 256, 256, 0, stream>>>(ncb, pred, maxv);
    rect_stats_kernel<<<K_, 256, 0, stream>>>(geo, labs, pred, maxv, out);
    mf_kernel        <<<K_, C_, 0, stream>>>(fm, geo, mf);
    pool_wmma_kernel <<<(K_ / 16) * 2, 32, 0, stream>>>(mf, cls_w, cls_b, out);
}